// MultiHeadAttention_22823456211846
// MI455X (gfx1250) — compile-verified
//
#include <hip/hip_runtime.h>
#include <hip/hip_bf16.h>
#include <math.h>

// ---------------------------------------------------------------------------
// MI455X (gfx1250) fused multi-head attention layer, fp32 via WMMA 16x16x4.
// QKV proj GEMMs -> fused scores/softmax/context (317KB LDS, async-LDS
// double-buffered staging) -> output proj + residual -> LayerNorm.
// ---------------------------------------------------------------------------

#define BATCH 4
#define SEQ 1024
#define DMODEL 1024
#define NHEADS 16
#define DK 64
#define DV 64

typedef __attribute__((ext_vector_type(2))) float v2f;
typedef __attribute__((ext_vector_type(8))) float v8f;
typedef int v4i __attribute__((vector_size(16)));   // matches builtin param

__device__ __forceinline__ v8f wmma4(v2f a, v2f b, v8f c) {
  return __builtin_amdgcn_wmma_f32_16x16x4_f32(
      false, a, false, b, (short)0, c, false, false);
}

// ---- CDNA5 async global->LDS copy (ASYNCcnt-tracked), guarded fallback ----
#if __has_builtin(__builtin_amdgcn_global_load_async_to_lds_b128) && \
    __has_builtin(__builtin_amdgcn_s_wait_asynccnt)
#define HAVE_ASYNC_LDS 1
#define ASYNC_WAIT(n) __builtin_amdgcn_s_wait_asynccnt(n)
#else
#define HAVE_ASYNC_LDS 0
#define ASYNC_WAIT(n)
#endif

__device__ __forceinline__ void cp16(float* lds, const float* g) {
#if HAVE_ASYNC_LDS
  __builtin_amdgcn_global_load_async_to_lds_b128(
      (__attribute__((address_space(1))) v4i*)g,
      (__attribute__((address_space(3))) v4i*)lds, 0, 0);
#else
  *(float4*)lds = *(const float4*)g;
#endif
}

// ---------------------------------------------------------------------------
// GEMM: [4096 x 1024] @ [1024 x 1024].  128x64 tile per 256-thread block,
// 8 waves, each wave owns 32x32 (4 WMMA accumulators).  Double-buffered
// async LDS staging of 128x32 X tile + 32x64 W tile.
// qkv_layout==1: out[((b*16+h)*1024+s)*64+d]  (per-head scatter)
// qkv_layout==0: out[m*1024+c] = acc + bias[c] + residual[m*1024+c]
// ---------------------------------------------------------------------------
#define TMG 128
#define TNG 64
#define TKC 32
#define AS_STRIDE 36   // 32 + pad (multiple of 4 floats -> 16B rows)
#define BS_STRIDE 68   // 64 + pad

__device__ __forceinline__ void gemm_stage(float* as, float* bs,
                                           const float* X, const float* W,
                                           int m0, int n0, int k0, int t) {
  const int xrow = t >> 1;          // 0..127, 2 threads/row, 16 floats each
  const int xc = (t & 1) * 16;
  const float* xg = &X[(size_t)(m0 + xrow) * DMODEL + k0 + xc];
  float* xl = &as[xrow * AS_STRIDE + xc];
  cp16(xl + 0, xg + 0);
  cp16(xl + 4, xg + 4);
  cp16(xl + 8, xg + 8);
  cp16(xl + 12, xg + 12);
  const int wrow = t >> 3;          // 0..31, 8 threads/row, 8 floats each
  const int wc = (t & 7) * 8;
  const float* wg = &W[(size_t)(k0 + wrow) * DMODEL + n0 + wc];
  float* wl = &bs[wrow * BS_STRIDE + wc];
  cp16(wl + 0, wg + 0);
  cp16(wl + 4, wg + 4);
}                                   // 6 async b128 per thread per stage

__global__ __launch_bounds__(256) void gemm_wmma_f32(
    const float* __restrict__ X, const float* __restrict__ W,
    const float* __restrict__ bias, const float* __restrict__ residual,
    float* __restrict__ out, int qkv_layout) {
  __shared__ float As[2][TMG * AS_STRIDE];
  __shared__ float Bs[2][TKC * BS_STRIDE];

  const int t = threadIdx.x;
  const int lane = t & 31;
  const int w = t >> 5;
  const int m0 = blockIdx.y * TMG;
  const int n0 = blockIdx.x * TNG;
  const int msub = (w & 3) * 32;            // 4 row-groups of 32
  const int nsub = (w >> 2) * 32;           // 2 col-groups of 32

  const int frow = msub + (lane & 15);
  const int fcol = lane & 15;
  const int klo = (lane >> 4) * 2;

  v8f acc00 = {}, acc01 = {}, acc10 = {}, acc11 = {};

  gemm_stage(As[0], Bs[0], X, W, m0, n0, 0, t);

  for (int k0 = 0; k0 < DMODEL; k0 += TKC) {
    const int buf = (k0 >> 5) & 1;
    if (k0 + TKC < DMODEL) {
      gemm_stage(As[buf ^ 1], Bs[buf ^ 1], X, W, m0, n0, k0 + TKC, t);
      ASYNC_WAIT(6);                        // retire current stage only
    } else {
      ASYNC_WAIT(0);
    }
    __syncthreads();
    const float* as = As[buf];
    const float* bs = Bs[buf];
#pragma unroll
    for (int kk = 0; kk < TKC; kk += 4) {
      const int kb = kk + klo;
      v2f a0, a1;
      a0.x = as[frow * AS_STRIDE + kb];
      a0.y = as[frow * AS_STRIDE + kb + 1];
      a1.x = as[(frow + 16) * AS_STRIDE + kb];
      a1.y = as[(frow + 16) * AS_STRIDE + kb + 1];
      v2f b0, b1;
      b0.x = bs[kb * BS_STRIDE + nsub + fcol];
      b0.y = bs[(kb + 1) * BS_STRIDE + nsub + fcol];
      b1.x = bs[kb * BS_STRIDE + nsub + 16 + fcol];
      b1.y = bs[(kb + 1) * BS_STRIDE + nsub + 16 + fcol];
      acc00 = wmma4(a0, b0, acc00);
      acc01 = wmma4(a0, b1, acc01);
      acc10 = wmma4(a1, b0, acc10);
      acc11 = wmma4(a1, b1, acc11);
    }
    __syncthreads();
  }

#pragma unroll
  for (int r = 0; r < 8; ++r) {
    const int gm0 = m0 + msub + r + 8 * (lane >> 4);
    const int gm1 = gm0 + 16;
    const int gn0 = n0 + nsub + fcol;
    const int gn1 = gn0 + 16;
    float v00 = acc00[r] + bias[gn0];
    float v01 = acc01[r] + bias[gn1];
    float v10 = acc10[r] + bias[gn0];
    float v11 = acc11[r] + bias[gn1];
    if (qkv_layout) {
      const int h0 = gn0 >> 6, d0 = gn0 & 63;
      const int h1 = gn1 >> 6, d1 = gn1 & 63;
      const int b0i = gm0 >> 10, s0 = gm0 & 1023;
      const int b1i = gm1 >> 10, s1 = gm1 & 1023;
      out[((size_t)(b0i * NHEADS + h0) * SEQ + s0) * DK + d0] = v00;
      out[((size_t)(b0i * NHEADS + h1) * SEQ + s0) * DK + d1] = v01;
      out[((size_t)(b1i * NHEADS + h0) * SEQ + s1) * DK + d0] = v10;
      out[((size_t)(b1i * NHEADS + h1) * SEQ + s1) * DK + d1] = v11;
    } else {
      out[(size_t)gm0 * DMODEL + gn0] = v00 + residual[(size_t)gm0 * DMODEL + gn0];
      out[(size_t)gm0 * DMODEL + gn1] = v01 + residual[(size_t)gm0 * DMODEL + gn1];
      out[(size_t)gm1 * DMODEL + gn0] = v10 + residual[(size_t)gm1 * DMODEL + gn0];
      out[(size_t)gm1 * DMODEL + gn1] = v11 + residual[(size_t)gm1 * DMODEL + gn1];
    }
  }
}

// ---------------------------------------------------------------------------
// Fused attention: one block per (b, h, 64-query block).  64x1024 score
// stripe in LDS; double-buffered async staging for K and V chunks; first V
// chunk prefetch overlaps the softmax pass.
// ---------------------------------------------------------------------------
#define SC_STRIDE 1032
#define QK_STRIDE 68

__device__ __forceinline__ void stage64(float* dst, const float* src, int t) {
  const int row = t >> 2, c0 = (t & 3) * 16;
  const float* g = &src[(size_t)row * 64 + c0];
  float* l = &dst[row * QK_STRIDE + c0];
  cp16(l + 0, g + 0);
  cp16(l + 4, g + 4);
  cp16(l + 8, g + 8);
  cp16(l + 12, g + 12);
}                                   // 4 async b128 per thread per stage

__global__ __launch_bounds__(256) void attn_fused(
    const float* __restrict__ qg, const float* __restrict__ kg,
    const float* __restrict__ vg, const unsigned char* __restrict__ mask,
    float* __restrict__ attn_out, float* __restrict__ ctx) {
  __shared__ float sc[64 * SC_STRIDE];      // 264KB score stripe
  __shared__ float qs[64 * QK_STRIDE];
  __shared__ float ks[2][64 * QK_STRIDE];   // K chunks, then V chunks
  __shared__ float red[256];

  const int t = threadIdx.x;
  const int lane = t & 31;
  const int w = t >> 5;
  const int qb = blockIdx.x & 15;
  const int bh = blockIdx.x >> 4;
  const int h = bh & 15;
  const int b = bh >> 4;
  const int q0 = qb * 64;
  const size_t head_off = (size_t)bh * SEQ * DK;

  const int msub = (w & 3) * 16;
  const int nsub = (w >> 2) * 32;
  const int frow = msub + (lane & 15);
  const int fcol = lane & 15;
  const int klo = (lane >> 4) * 2;

  stage64(qs, &qg[head_off + (size_t)q0 * DK], t);
  stage64(ks[0], &kg[head_off], t);

  // --- scores = q k^T / 8, masked fill -1e9 ---
  for (int kc = 0; kc < 16; ++kc) {
    const int buf = kc & 1;
    if (kc < 15) {
      stage64(ks[buf ^ 1], &kg[head_off + (size_t)(kc + 1) * 64 * DK], t);
      ASYNC_WAIT(4);
    } else {
      ASYNC_WAIT(0);
    }
    __syncthreads();
    const float* kb_lds = ks[buf];

    v8f acc0 = {}, acc1 = {};
#pragma unroll
    for (int kk = 0; kk < DK; kk += 4) {
      const int kb = kk + klo;
      v2f a;
      a.x = qs[frow * QK_STRIDE + kb];
      a.y = qs[frow * QK_STRIDE + kb + 1];
      v2f b0, b1;   // B[kk][n] = k[key=n][kk]: contiguous pair per key row
      b0.x = kb_lds[(nsub + fcol) * QK_STRIDE + kb];
      b0.y = kb_lds[(nsub + fcol) * QK_STRIDE + kb + 1];
      b1.x = kb_lds[(nsub + 16 + fcol) * QK_STRIDE + kb];
      b1.y = kb_lds[(nsub + 16 + fcol) * QK_STRIDE + kb + 1];
      acc0 = wmma4(a, b0, acc0);
      acc1 = wmma4(a, b1, acc1);
    }
#pragma unroll
    for (int r = 0; r < 8; ++r) {
      const int lr = msub + r + 8 * (lane >> 4);
      const int qgl = q0 + lr;
      const int c0g = kc * 64 + nsub + fcol;
      const unsigned char mk0 = mask[((size_t)b * SEQ + qgl) * SEQ + c0g];
      const unsigned char mk1 = mask[((size_t)b * SEQ + qgl) * SEQ + c0g + 16];
      sc[lr * SC_STRIDE + c0g] = mk0 ? -1e9f : acc0[r] * 0.125f;
      sc[lr * SC_STRIDE + c0g + 16] = mk1 ? -1e9f : acc1[r] * 0.125f;
    }
    __syncthreads();
  }

  stage64(ks[0], &vg[head_off], t);  // V chunk 0 hides under softmax

  // --- softmax over 1024 cols: 4 threads per row, float4 vectorized ---
  {
    const int r = t >> 2, part = t & 3;
    float4* srow4 = (float4*)&sc[r * SC_STRIDE];
    float pm = -3.4e38f;
    for (int c4 = part * 64; c4 < part * 64 + 64; ++c4) {
      const float4 v = srow4[c4];
      pm = fmaxf(fmaxf(pm, fmaxf(v.x, v.y)), fmaxf(v.z, v.w));
    }
    red[t] = pm;
    __syncthreads();
    const float rmax = fmaxf(fmaxf(red[r * 4 + 0], red[r * 4 + 1]),
                             fmaxf(red[r * 4 + 2], red[r * 4 + 3]));
    __syncthreads();
    float ps = 0.f;
    for (int c4 = part * 64; c4 < part * 64 + 64; ++c4) {
      float4 v = srow4[c4];
      v.x = __expf(v.x - rmax);
      v.y = __expf(v.y - rmax);
      v.z = __expf(v.z - rmax);
      v.w = __expf(v.w - rmax);
      srow4[c4] = v;
      ps += v.x + v.y + v.z + v.w;
    }
    red[t] = ps;
    __syncthreads();
    const float inv = 1.f / (red[r * 4 + 0] + red[r * 4 + 1] +
                             red[r * 4 + 2] + red[r * 4 + 3]);
    float4* aout4 = (float4*)&attn_out[((size_t)bh * SEQ + (q0 + r)) * SEQ];
    for (int c4 = part * 64; c4 < part * 64 + 64; ++c4) {
      float4 v = srow4[c4];
      v.x *= inv; v.y *= inv; v.z *= inv; v.w *= inv;
      srow4[c4] = v;          // normalized probs for context GEMM
      aout4[c4] = v;          // required attn output (HBM stream)
    }
  }

  // --- context = P(64x1024) @ V(1024x64) ---
  v8f cacc0 = {}, cacc1 = {};
  for (int kc = 0; kc < 16; ++kc) {
    const int buf = kc & 1;
    if (kc < 15) {
      stage64(ks[buf ^ 1], &vg[head_off + (size_t)(kc + 1) * 64 * DV], t);
      ASYNC_WAIT(4);
    } else {
      ASYNC_WAIT(0);
    }
    __syncthreads();
    const float* vb_lds = ks[buf];
#pragma unroll
    for (int kk = 0; kk < 64; kk += 4) {
      const int kb = kk + klo;
      v2f a;
      a.x = sc[frow * SC_STRIDE + kc * 64 + kb];
      a.y = sc[frow * SC_STRIDE + kc * 64 + kb + 1];
      v2f b0, b1;   // B[kk][d] = v[key=kb..][d]
      b0.x = vb_lds[kb * QK_STRIDE + nsub + fcol];
      b0.y = vb_lds[(kb + 1) * QK_STRIDE + nsub + fcol];
      b1.x = vb_lds[kb * QK_STRIDE + nsub + 16 + fcol];
      b1.y = vb_lds[(kb + 1) * QK_STRIDE + nsub + 16 + fcol];
      cacc0 = wmma4(a, b0, cacc0);
      cacc1 = wmma4(a, b1, cacc1);
    }
    __syncthreads();
  }
#pragma unroll
  for (int r = 0; r < 8; ++r) {
    const int lr = msub + r + 8 * (lane >> 4);
    const int sq = q0 + lr;
    const int d0 = nsub + fcol;
    ctx[((size_t)b * SEQ + sq) * DMODEL + h * DV + d0] = cacc0[r];
    ctx[((size_t)b * SEQ + sq) * DMODEL + h * DV + d0 + 16] = cacc1[r];
  }
}

// ---------------------------------------------------------------------------
// Row LayerNorm, in place on y = ctx@Wo + bo + Q.  One block per row.
// ---------------------------------------------------------------------------
__global__ __launch_bounds__(256) void layernorm_rows(
    float* __restrict__ y, const float* __restrict__ gamma,
    const float* __restrict__ beta) {
  __shared__ float rsum[256];
  __shared__ float rsq[256];
  const int t = threadIdx.x;
  float* p = &y[(size_t)blockIdx.x * DMODEL];
  const float4 v = ((const float4*)p)[t];
  rsum[t] = v.x + v.y + v.z + v.w;
  rsq[t] = v.x * v.x + v.y * v.y + v.z * v.z + v.w * v.w;
  __syncthreads();
  for (int off = 128; off > 0; off >>= 1) {
    if (t < off) {
      rsum[t] += rsum[t + off];
      rsq[t] += rsq[t + off];
    }
    __syncthreads();
  }
  const float mean = rsum[0] * (1.f / DMODEL);
  const float var = rsq[0] * (1.f / DMODEL) - mean * mean;
  const float inv = rsqrtf(var + 1e-5f);
  const float4 g = ((const float4*)gamma)[t];
  const float4 bt = ((const float4*)beta)[t];
  float4 o;
  o.x = (v.x - mean) * inv * g.x + bt.x;
  o.y = (v.y - mean) * inv * g.y + bt.y;
  o.z = (v.z - mean) * inv * g.z + bt.z;
  o.w = (v.w - mean) * inv * g.w + bt.w;
  ((float4*)p)[t] = o;
}

// ---------------------------------------------------------------------------
extern "C" void kernel_launch(void* const* d_in, const int* in_sizes, int n_in,
                              void* d_out, int out_size, void* d_ws,
                              size_t ws_size, hipStream_t stream) {
  const float* Q = (const float*)d_in[0];
  const float* K = (const float*)d_in[1];
  const float* V = (const float*)d_in[2];
  const unsigned char* mask = (const unsigned char*)d_in[3];  // bool mask
  const float* Wq = (const float*)d_in[4];
  const float* bq = (const float*)d_in[5];
  const float* Wk = (const float*)d_in[6];
  const float* bk = (const float*)d_in[7];
  const float* Wv = (const float*)d_in[8];
  const float* bv = (const float*)d_in[9];
  const float* Wo = (const float*)d_in[10];
  const float* bo = (const float*)d_in[11];
  const float* gamma = (const float*)d_in[12];
  const float* beta = (const float*)d_in[13];

  float* out = (float*)d_out;                              // [B,S,DMODEL]
  float* attn = out + (size_t)BATCH * SEQ * DMODEL;        // [B,H,S,S]

  float* ws = (float*)d_ws;                                // 64 MB used
  float* qh = ws;                                          // [B,H,S,DK]
  float* kh = ws + (size_t)BATCH * SEQ * DMODEL;
  float* vh = ws + 2 * (size_t)BATCH * SEQ * DMODEL;
  float* ctx = ws + 3 * (size_t)BATCH * SEQ * DMODEL;      // [B,S,H*DV]

  const dim3 gg(DMODEL / TNG, (BATCH * SEQ) / TMG);        // (16, 32)
  const dim3 blk(256);
  gemm_wmma_f32<<<gg, blk, 0, stream>>>(Q, Wq, bq, nullptr, qh, 1);
  gemm_wmma_f32<<<gg, blk, 0, stream>>>(K, Wk, bk, nullptr, kh, 1);
  gemm_wmma_f32<<<gg, blk, 0, stream>>>(V, Wv, bv, nullptr, vh, 1);

  attn_fused<<<BATCH * NHEADS * (SEQ / 64), blk, 0, stream>>>(qh, kh, vh, mask,
                                                              attn, ctx);

  gemm_wmma_f32<<<gg, blk, 0, stream>>>(ctx, Wo, bo, Q, out, 0);
  layernorm_rows<<<BATCH * SEQ, blk, 0, stream>>>(out, gamma, beta);
}